// SupConLossDDP_70102456205719
// MI455X (gfx1250) — compile-verified
//
#include <hip/hip_runtime.h>
#include <math.h>

// ---- problem constants (from reference) ----
#define BSZ      2048
#define NVIEWS   2
#define DIM      128
#define NTOT     4096            // BSZ*NVIEWS
#define INV_T    2.0f            // 1/TEMPERATURE
#define RATIO    (0.5f/0.07f)    // TEMPERATURE/BASE_TEMPERATURE
#define ROWS_PER_BLOCK 16
#define THREADS  256             // 8 wave32 waves

typedef __attribute__((ext_vector_type(2))) float v2f;
typedef __attribute__((ext_vector_type(8))) float v8f;

// CF[r][k] = F[(r%2048)*256 + (r/2048)*128 + k]  (view-major permutation)
__device__ __forceinline__ size_t cf_base(int r) {
    return (size_t)(r & (BSZ - 1)) * (NVIEWS * DIM) + (size_t)(r >> 11) * DIM;
}

__device__ __forceinline__ float wave_max(float v) {
    #pragma unroll
    for (int o = 16; o > 0; o >>= 1) v = fmaxf(v, __shfl_xor(v, o, 32));
    return v;
}
__device__ __forceinline__ float wave_sum(float v) {
    #pragma unroll
    for (int o = 16; o > 0; o >>= 1) v += __shfl_xor(v, o, 32);
    return v;
}

// One workgroup per 16-row strip: WMMA-f32 the 16x4096 strip of the Gram
// matrix into LDS (256 KB, a CDNA5 320KB-WGP capability), then softmax
// stats + outputs straight from LDS — S never round-trips through HBM.
__global__ void __launch_bounds__(THREADS)
supcon_gram_kernel(const float* __restrict__ F,
                   const int*   __restrict__ labels,
                   float*       __restrict__ out,      // d_out
                   float*       __restrict__ ws)       // per-row mean_log_prob_pos
{
    extern __shared__ float smem[];                    // [16*4096] S strip
    int* slab = (int*)(smem + ROWS_PER_BLOCK * NTOT);  // [2048] labels

    const int tid  = threadIdx.x;
    const int lane = tid & 31;
    const int wv   = tid >> 5;          // 0..7
    const int row0 = blockIdx.x * ROWS_PER_BLOCK;

    for (int j = tid; j < BSZ; j += THREADS) slab[j] = labels[j];

    // ---- phase 1: Gram strip via V_WMMA_F32_16X16X4_F32 ----
    const int m    = lane & 15;
    const int half = lane >> 4;
    const size_t abase = cf_base(row0 + m);

    v2f afrag[32];                       // 16x128 A block, K-chunks of 4
    #pragma unroll
    for (int kk = 0; kk < 32; ++kk)
        afrag[kk] = *(const v2f*)(F + abase + kk * 4 + half * 2);

    for (int ct = 0; ct < 32; ++ct) {    // 256 col-tiles / 8 waves
        const int col0 = (ct * 8 + wv) * 16;
        const size_t bbase = cf_base(col0 + m);
        v8f acc = {};
        #pragma unroll
        for (int kk = 0; kk < 32; ++kk) {
            v2f b = *(const v2f*)(F + bbase + kk * 4 + half * 2);
            acc = __builtin_amdgcn_wmma_f32_16x16x4_f32(
                      false, afrag[kk], false, b, (short)0, acc, false, false);
        }
        #pragma unroll
        for (int v = 0; v < 8; ++v) {
            const int mr = v + half * 8;                 // C/D layout
            smem[mr * NTOT + col0 + m] = acc[v] * INV_T; // logits pre-shift
        }
    }
    __syncthreads();

    // ---- phase 2: per-row softmax stats + clean_logits, from LDS ----
    float* clean = out + 1;
    #pragma unroll
    for (int rr = 0; rr < 2; ++rr) {
        const int r  = wv * 2 + rr;          // 16 rows / 8 waves
        const int gi = row0 + r;
        const float* row = smem + r * NTOT;
        const int mylab = slab[gi & (BSZ - 1)];

        // row max (diagonal included, as in reference)
        float mx = -INFINITY;
        #pragma unroll 4
        for (int j = lane; j < NTOT; j += 32) mx = fmaxf(mx, row[j]);
        mx = wave_max(mx);

        // denom = sum_{j!=i} exp(s-mx); possum/cnt over same-label off-diag
        float denom = 0.f, possum = 0.f, cnt = 0.f;
        #pragma unroll 4
        for (int j = lane; j < NTOT; j += 32) {
            if (j == gi) continue;
            const float sh = row[j] - mx;
            denom += __expf(sh);
            if (slab[j & (BSZ - 1)] == mylab) { possum += sh; cnt += 1.f; }
        }
        denom  = wave_sum(denom);
        possum = wave_sum(possum);
        cnt    = wave_sum(cnt);

        if (lane == 0)
            ws[gi] = possum / cnt - logf(denom);   // mean_log_prob_pos[i]

        // clean_logits row: exp(s-mx) with diagonal column removed
        const size_t obase = (size_t)gi * (NTOT - 1);
        #pragma unroll 4
        for (int j = lane; j < NTOT - 1; j += 32) {
            const int c = j + (j >= gi);
            clean[obase + j] = __expf(row[c] - mx);
        }
    }
}

// Deterministic fixed-tree loss reduction + label tail.
__global__ void __launch_bounds__(THREADS)
supcon_finalize_kernel(const float* __restrict__ ws, float* __restrict__ out)
{
    __shared__ float red[THREADS];
    const int tid = threadIdx.x;

    if (blockIdx.x == 0) {
        float s = 0.f;
        for (int i = tid; i < NTOT; i += THREADS) s += ws[i];
        red[tid] = s;
        __syncthreads();
        for (int o = THREADS / 2; o > 0; o >>= 1) {
            if (tid < o) red[tid] += red[tid + o];
            __syncthreads();
        }
        if (tid == 0) out[0] = -RATIO * red[0] / (float)NTOT;
    }

    // out_labels = concat(arange(2048)+2047, arange(2048)), stored as float
    const int t = blockIdx.x * THREADS + tid;   // grid of 16 blocks covers 4096
    if (t < NTOT) {
        const float lv = (t < BSZ) ? (float)(t + BSZ - 1) : (float)(t - BSZ);
        out[(size_t)1 + (size_t)NTOT * (NTOT - 1) + t] = lv;
    }
}

extern "C" void kernel_launch(void* const* d_in, const int* in_sizes, int n_in,
                              void* d_out, int out_size, void* d_ws, size_t ws_size,
                              hipStream_t stream) {
    const float* F      = (const float*)d_in[0];
    const int*   labels = (const int*)d_in[1];
    float* out = (float*)d_out;
    float* ws  = (float*)d_ws;   // 4096 floats of scratch

    const size_t shmem = (size_t)ROWS_PER_BLOCK * NTOT * sizeof(float)  // 256 KB S strip
                       + (size_t)BSZ * sizeof(int);                     // 8 KB labels
    supcon_gram_kernel<<<NTOT / ROWS_PER_BLOCK, THREADS, shmem, stream>>>(F, labels, out, ws);
    supcon_finalize_kernel<<<NTOT / THREADS, THREADS, 0, stream>>>(ws, out);
}